// MutiltaskGcnNet_8710193676879
// MI455X (gfx1250) — compile-verified
//
#include <hip/hip_runtime.h>
#include <hip/hip_bf16.h>

typedef float v2f __attribute__((ext_vector_type(2)));
typedef float v8f __attribute__((ext_vector_type(8)));
typedef unsigned int u32x4 __attribute__((ext_vector_type(4)));
typedef int i32x4 __attribute__((ext_vector_type(4)));
typedef int i32x8 __attribute__((ext_vector_type(8)));

__device__ __forceinline__ unsigned hash_u32(unsigned x) {
    x ^= x >> 16; x *= 0x7feb352dU;
    x ^= x >> 15; x *= 0x846ca68bU;
    x ^= x >> 16;
    return x;
}

// ---------------------------------------------------------------------------
// TDM: DMA a 2-D fp32 panel [rows x cols] (row-major, stride == cols) from
// global memory into LDS at byte offset lds_byte_addr. Descriptor per CDNA5
// D# spec (CDNA5 ISA). Groups 2/3 zero (<=2D tile, dims unused).
// This toolchain exposes the 6-arg clang-23 builtin:
//   (u32x4 g0, i32x8 g1, i32x4 g2, i32x4 g3, i32x8 extra, i32 cpol)
// ---------------------------------------------------------------------------
__device__ __forceinline__ void tdm_load_panel(const float* gsrc, unsigned lds_byte_addr,
                                               int rows, int cols) {
    unsigned long long ga = (unsigned long long)(size_t)gsrc;
    u32x4 g0;
    g0.x = 1u;                                      // count=1, user descriptor
    g0.y = lds_byte_addr;                           // lds_addr (bytes)
    g0.z = (unsigned)(ga & 0xFFFFFFFFu);            // global_addr[31:0]
    g0.w = (unsigned)((ga >> 32) & 0x1FFFFFFu)      // global_addr[56:32]
         | (2u << 30);                              // type=2 ("image")
    i32x8 g1;
    g1[0] = (2 << 16);                              // wg_mask=0, data_size=2 (4B)
    g1[1] = (int)((unsigned)(cols & 0xFFFF) << 16); // tensor_dim0[15:0] (abar addr=0)
    g1[2] = (int)(((unsigned)cols >> 16) | ((unsigned)(rows & 0xFFFF) << 16)); // dim0 hi | dim1 lo
    g1[3] = (int)((unsigned)(cols & 0xFFFF) << 16); // dim1 hi=0 | tile_dim0
    g1[4] = (int)(rows & 0xFFFF);                   // tile_dim1 | tile_dim2=0
    g1[5] = cols;                                   // tensor_dim0_stride[31:0]
    g1[6] = 0;                                      // stride0 hi | stride1 lo
    g1[7] = 0;
    i32x4 z4 = {0, 0, 0, 0};
    i32x8 z8 = {0, 0, 0, 0, 0, 0, 0, 0};
    __builtin_amdgcn_tensor_load_to_lds(g0, g1, z4, z4, z8, 0);
}

// dst[i] = bias[i % cols]  (pre-loads bias so SpMM accumulates straight onto it)
__global__ void bcast_bias_kernel(float* __restrict__ dst, const float* __restrict__ bias,
                                  int cols, long long total) {
    long long i = (long long)blockIdx.x * blockDim.x + threadIdx.x;
    if (i >= total) return;
    dst[i] = bias[(int)(i % cols)];
}

// ---------------------------------------------------------------------------
// GEMM1: C[M,256] = A[M,K] @ B[K,256], fp32 WMMA 16x16x4, B staged into LDS
// by the Tensor Data Mover (double-buffered K-panels of KT rows), overlapped
// with compute via TENSORcnt. Block = 128 threads (4 waves x 16x64 strips).
// ---------------------------------------------------------------------------
template <int KT>
__global__ void gemm1_wmma_tdm_kernel(const float* __restrict__ A, const float* __restrict__ B,
                                      float* __restrict__ C, int M, int K) {
    __shared__ float bpanel[2][KT * 256];

    const int lane = threadIdx.x & 31;
    const int wave = threadIdx.x >> 5;
    const int m0   = blockIdx.x * 16;
    const int n0   = wave * 64;
    const int koff = (lane >> 4) << 1;   // lanes 0-15: K+{0,1}; lanes 16-31: K+{2,3}
    const int half = lane & 15;

    int mr = m0 + half;                  // clamp; no divergence, EXEC stays all-1 for WMMA
    if (mr >= M) mr = M - 1;
    const float* __restrict__ Arow = A + (long long)mr * K + koff;

    const unsigned lds0 = (unsigned)(size_t)&bpanel[0][0];
    const unsigned lds1 = (unsigned)(size_t)&bpanel[1][0];

    v8f acc[4] = {};

    const int npanel = K / KT;
    if (wave == 0) {
        tdm_load_panel(B, lds0, KT, 256);
        __builtin_amdgcn_s_wait_tensorcnt(0);
    }
    __syncthreads();

    for (int p = 0; p < npanel; ++p) {
        if (wave == 0 && (p + 1) < npanel)   // prefetch next panel while computing this one
            tdm_load_panel(B + (long long)(p + 1) * KT * 256,
                           ((p + 1) & 1) ? lds1 : lds0, KT, 256);

        const float* __restrict__ Bp = bpanel[p & 1];
        const float* __restrict__ Ap = Arow + p * KT;
#pragma unroll
        for (int kk = 0; kk < KT; kk += 4) {
            v2f a = *(const v2f*)(Ap + kk);          // 8B aligned: kk%4==0, koff even
            const float* __restrict__ Bk = Bp + (kk + koff) * 256;
#pragma unroll
            for (int t = 0; t < 4; ++t) {
                const int col = n0 + t * 16 + half;
                v2f b;
                b.x = Bk[col];                       // ds_load, conflict-free (col = lane)
                b.y = Bk[256 + col];
                acc[t] = __builtin_amdgcn_wmma_f32_16x16x4_f32(
                    false, a, false, b, (short)0, acc[t], false, false);
            }
        }

        if (wave == 0) __builtin_amdgcn_s_wait_tensorcnt(0);
        __syncthreads();
    }

    // C/D layout: VGPR r -> row m0+r (lanes 0-15) / m0+8+r (lanes 16-31)
    const int rbase = m0 + ((lane >> 4) << 3);
#pragma unroll
    for (int t = 0; t < 4; ++t) {
#pragma unroll
        for (int r = 0; r < 8; ++r) {
            const int row = rbase + r;
            if (row < M)
                C[(long long)row * 256 + n0 + t * 16 + half] = acc[t][r];
        }
    }
}

// ---------------------------------------------------------------------------
// Small GEMM: C[M,Nout] = A[M,K] @ B[K,Nout]; one wave per 16 x (NT*16) strip,
// B read straight from L2 (W21/W22 are 64KB/16KB, fully cache-resident).
// ---------------------------------------------------------------------------
template <int NT>
__global__ void gemm_wmma_kernel(const float* __restrict__ A, const float* __restrict__ B,
                                 float* __restrict__ C, int M, int K, int Nout) {
    const int lane = threadIdx.x & 31;
    const int wave = threadIdx.x >> 5;
    const int m0   = blockIdx.x * 16;
    const int n0   = wave * (NT * 16);
    const int koff = (lane >> 4) << 1;
    const int half = lane & 15;

    int mr = m0 + half;
    if (mr >= M) mr = M - 1;
    const float* __restrict__ Arow = A + (long long)mr * K + koff;

    v8f acc[NT] = {};

    for (int k = 0; k < K; k += 4) {
        v2f a = *(const v2f*)(Arow + k);
        const float* __restrict__ Bk = B + (long long)(k + koff) * Nout;
#pragma unroll
        for (int t = 0; t < NT; ++t) {
            const int col = n0 + t * 16 + half;
            v2f b;
            b.x = Bk[col];
            b.y = Bk[Nout + col];
            acc[t] = __builtin_amdgcn_wmma_f32_16x16x4_f32(
                false, a, false, b, (short)0, acc[t], false, false);
        }
    }

    const int rbase = m0 + ((lane >> 4) << 3);
#pragma unroll
    for (int t = 0; t < NT; ++t) {
#pragma unroll
        for (int r = 0; r < 8; ++r) {
            const int row = rbase + r;
            if (row < M)
                C[(long long)row * Nout + n0 + t * 16 + half] = acc[t][r];
        }
    }
}

// Y[rows[e], :] += vals[e] * X[cols[e], :] — one wave per edge, lane-strided over D.
// X (N*256*4B = 51MB) is L2-resident on MI455X (192MB L2): gathers/atomics hit L2.
__global__ void spmm_atomic_kernel(const int* __restrict__ rows, const int* __restrict__ cols,
                                   const float* __restrict__ vals,
                                   const float* __restrict__ X, float* __restrict__ Y,
                                   int D, int E) {
    const int gid  = blockIdx.x * blockDim.x + threadIdx.x;
    const int e    = gid >> 5;
    const int lane = gid & 31;
    if (e >= E) return;
    const int r = rows[e];
    const int c = cols[e];
    const float v = vals[e];
    const float* __restrict__ xr = X + (long long)c * D;
    float* __restrict__ yr = Y + (long long)r * D;
    for (int d = lane; d < D; d += 32)
        atomicAdd(yr + d, v * xr[d]);
}

// share = relu(H); H = dropout(share) in place (p=0.5 -> one hash bit, x2 scale)
__global__ void relu_dropout_kernel(float* __restrict__ H, float* __restrict__ share,
                                    long long total) {
    long long i = (long long)blockIdx.x * blockDim.x + threadIdx.x;
    if (i >= total) return;
    const float h = fmaxf(H[i], 0.0f);
    share[i] = h;
    const unsigned keep = hash_u32((unsigned)i ^ 0x9E3779B9u) & 1u;
    H[i] = keep ? (h * 2.0f) : 0.0f;
}

extern "C" void kernel_launch(void* const* d_in, const int* in_sizes, int n_in,
                              void* d_out, int out_size, void* d_ws, size_t ws_size,
                              hipStream_t stream) {
    (void)n_in; (void)out_size; (void)ws_size;

    const int*   rows = (const int*)d_in[0];
    const int*   cols = (const int*)d_in[1];
    const float* vals = (const float*)d_in[2];
    const float* feat = (const float*)d_in[3];
    const float* W1   = (const float*)d_in[4];
    const float* b1   = (const float*)d_in[5];
    const float* W21  = (const float*)d_in[6];
    const float* b21  = (const float*)d_in[7];
    const float* W22  = (const float*)d_in[8];
    const float* b22  = (const float*)d_in[9];

    const int D0 = 512, D1 = 256, Da = 64, Db = 16;
    const int E  = in_sizes[0];
    const int N  = in_sizes[3] / D0;
    const int mt = (N + 15) / 16;          // 3125 M-tiles (N=50000 divides by 16)

    // Workspace layout (floats): [XW: N*256][H: N*256]; XW region reused for Z21/Z22.
    float* ws  = (float*)d_ws;
    float* XW  = ws;
    float* H   = ws + (long long)N * D1;
    float* Z21 = XW;
    float* Z22 = XW + (long long)N * Da;

    // Output layout: [logits2_1 N*64 | logits2_2 N*16 | logits_share N*256]
    float* out   = (float*)d_out;
    float* out21 = out;
    float* out22 = out + (long long)N * Da;
    float* outsh = out + (long long)N * (Da + Db);

    const long long t1 = (long long)N * D1;
    const long long ta = (long long)N * Da;
    const long long tb = (long long)N * Db;

    // Pre-fill accumulators with biases (fuses the "+b" of each GCN layer).
    bcast_bias_kernel<<<(unsigned)((t1 + 255) / 256), 256, 0, stream>>>(H,     b1,  D1, t1);
    bcast_bias_kernel<<<(unsigned)((ta + 255) / 256), 256, 0, stream>>>(out21, b21, Da, ta);
    bcast_bias_kernel<<<(unsigned)((tb + 255) / 256), 256, 0, stream>>>(out22, b22, Db, tb);

    // XW = feature @ W1 (K=512, Nout=256): TDM-staged B panels, 4 waves/block.
    gemm1_wmma_tdm_kernel<32><<<dim3(mt), dim3(128), 0, stream>>>(feat, W1, XW, N, D0);

    // H += A @ XW  (accumulates onto b1)
    const int spblocks = (int)(((long long)E * 32 + 255) / 256);
    spmm_atomic_kernel<<<spblocks, 256, 0, stream>>>(rows, cols, vals, XW, H, D1, E);

    // logits_share = relu(H); H = dropout(relu(H)) in place
    relu_dropout_kernel<<<(unsigned)((t1 + 255) / 256), 256, 0, stream>>>(H, outsh, t1);

    // Z21 = H @ W21 (Nout=64: one wave, NT=4); Z22 = H @ W22 (Nout=16: one wave, NT=1)
    gemm_wmma_kernel<4><<<dim3(mt), dim3(32), 0, stream>>>(H, W21, Z21, N, D1, Da);
    gemm_wmma_kernel<1><<<dim3(mt), dim3(32), 0, stream>>>(H, W22, Z22, N, D1, Db);

    // logits2_* = A @ Z2* + b2*  (accumulate straight into d_out, bias pre-loaded)
    spmm_atomic_kernel<<<spblocks, 256, 0, stream>>>(rows, cols, vals, Z21, out21, Da, E);
    spmm_atomic_kernel<<<spblocks, 256, 0, stream>>>(rows, cols, vals, Z22, out22, Db, E);
}